// TransformerLayer_11871289606898
// MI455X (gfx1250) — compile-verified
//
#include <hip/hip_runtime.h>
#include <math.h>

// ---------------------------------------------------------------------------
// TransformerLayer on MI455X (gfx1250, wave32).
// fp32 end-to-end via V_WMMA_F32_16X16X4_F32; attention K/V tiles staged into
// LDS with the Tensor Data Mover (double-buffered, TENSORcnt-synchronized).
// ---------------------------------------------------------------------------

typedef __attribute__((ext_vector_type(2))) float v2f;
typedef __attribute__((ext_vector_type(8))) float v8f;
typedef __attribute__((ext_vector_type(4))) unsigned int v4u;
typedef __attribute__((ext_vector_type(4))) int v4i;
typedef __attribute__((ext_vector_type(8))) int v8i;

#define TL_B  2
#define TL_L  2048
#define TL_S  2048
#define TL_D  256
#define TL_H  8
#define TL_HD 32
#define TL_EPS 1e-5f

// attention LDS layout (floats): scores + qtile + oacc + red + stat + staging
#define TL_SMEM_FLOATS (16 * TL_S + 512 + 512 + 128 + 16 + 4 * 2 * 512)

#if defined(__has_builtin)
#if __has_builtin(__builtin_amdgcn_tensor_load_to_lds)
#define TL_HAS_TDM 1
#endif
#if __has_builtin(__builtin_amdgcn_s_wait_tensorcnt)
#define TL_HAS_WAIT_TENSOR 1
#endif
#endif

__device__ __forceinline__ v8f wmma_f32(v2f a, v2f b, v8f c) {
  // 8 args: (neg_a, A, neg_b, B, c_mod, C, reuse_a, reuse_b)
  return __builtin_amdgcn_wmma_f32_16x16x4_f32(false, a, false, b, (short)0, c,
                                               false, false);
}

__device__ __forceinline__ void tl_wait_tensorcnt0() {
#ifdef TL_HAS_WAIT_TENSOR
  __builtin_amdgcn_s_wait_tensorcnt(0);
#else
  asm volatile("s_wait_tensorcnt 0x0" ::: "memory");
#endif
  asm volatile("" ::: "memory");  // keep LDS reads below the wait
}
__device__ __forceinline__ void tl_wait_tensorcnt1() {
#ifdef TL_HAS_WAIT_TENSOR
  __builtin_amdgcn_s_wait_tensorcnt(1);
#else
  asm volatile("s_wait_tensorcnt 0x1" ::: "memory");
#endif
  asm volatile("" ::: "memory");
}

#ifdef TL_HAS_TDM
// TDM: async-DMA a 16x32 fp32 tile (row stride TL_D elements) into LDS.
// D# built per CDNA5 ISA; operands forced wave-uniform.
__device__ __forceinline__ void tdm_load_tile_16x32(float* lds_ptr,
                                                    const float* gptr) {
  const uint32_t ldsa =
      (uint32_t)__builtin_amdgcn_readfirstlane((int)(uintptr_t)lds_ptr);
  const uint64_t ga = (uint64_t)(uintptr_t)gptr;
  const uint32_t galo = (uint32_t)__builtin_amdgcn_readfirstlane((int)(uint32_t)ga);
  const uint32_t gahi =
      (uint32_t)__builtin_amdgcn_readfirstlane((int)(uint32_t)(ga >> 32));
  v4u g0;
  g0.x = 1u;                                        // count=1, user mode
  g0.y = ldsa;                                      // lds_addr
  g0.z = galo;                                      // global_addr[31:0]
  g0.w = (gahi & 0x01FFFFFFu) | (2u << 30);         // addr[56:32] | type=2
  v8i g1;
  g1[0] = (int)(2u << 16);   // workgroup_mask=0, data_size=4B (code 2)
  g1[1] = (int)(32u << 16);  // tensor_dim0 = 32 (bits 79:48)
  g1[2] = (int)(16u << 16);  // tensor_dim1 = 16 (bits 111:80)
  g1[3] = (int)(32u << 16);  // tile_dim0 = 32   (bits 127:112)
  g1[4] = 16;                // tile_dim1 = 16, tile_dim2 = 0
  g1[5] = TL_D;              // tensor_dim0_stride = 256 (bits 207:160)
  g1[6] = 0;                 // tensor_dim1_stride (unused, 2D)
  g1[7] = 0;
  const v4i gz = {0, 0, 0, 0};  // groups 2/3 unused for 2D tiles
#if __clang_major__ >= 23
  const v8i gz8 = {0, 0, 0, 0, 0, 0, 0, 0};
  __builtin_amdgcn_tensor_load_to_lds(g0, g1, gz, gz, gz8, 0);
#else
  __builtin_amdgcn_tensor_load_to_lds(g0, g1, gz, gz, 0);
#endif
}
#endif  // TL_HAS_TDM

// ---------------------------------------------------------------------------
// C[M,N] = act( concat(A0,A1)[M,K] @ W[N,K]^T ).
// One wave computes a 16x64 strip (4 N-tiles): the A fragment is reused by
// 4 WMMAs per K-step (1 A load + 4 B loads -> 4 wmma), 4 independent
// accumulator chains hide global-load latency.
// ---------------------------------------------------------------------------
__global__ void gemm_nt_kernel(const float* __restrict__ A0,
                               const float* __restrict__ A1,
                               const float* __restrict__ W,
                               float* __restrict__ C,
                               int N, int K, int Ka, int relu) {
  const int lane = threadIdx.x & 31;
  const int wv   = threadIdx.x >> 5;
  const int lm   = lane & 15;          // A row (M) / B col (N) within tile
  const int kk   = (lane >> 4) << 1;   // K sub-offset: lanes 16-31 hold K+2
  const int tileN0 = blockIdx.x << 6;  // 64 N-cols per block/wave
  const int tileM  = ((blockIdx.y << 2) + wv) << 4;
  const int Kb = K - Ka;

  const size_t arow  = (size_t)(tileM + lm);
  const size_t wbase = (size_t)(tileN0 + lm) * (size_t)K;

  // warm the W rows this wave will stream (global_prefetch_b8)
  __builtin_prefetch(W + wbase, 0, 0);
  __builtin_prefetch(W + wbase + (size_t)16 * K, 0, 0);
  __builtin_prefetch(W + wbase + (size_t)32 * K, 0, 0);
  __builtin_prefetch(W + wbase + (size_t)48 * K, 0, 0);

  v8f acc0 = {}, acc1 = {}, acc2 = {}, acc3 = {};
#pragma unroll 4
  for (int k = 0; k < K; k += 4) {
    const int ka = k + kk;
    v2f a;
    if (ka < Ka) {                     // ka even, Ka even -> pair never splits
      const float* p = A0 + arow * (size_t)Ka + ka;
      a.x = p[0]; a.y = p[1];
    } else {
      const float* p = A1 + arow * (size_t)Kb + (ka - Ka);
      a.x = p[0]; a.y = p[1];
    }
    const float* q0 = W + wbase + ka;  // B(k,n) = W[n][k], contiguous in k
    v2f b0, b1, b2, b3;
    b0.x = q0[0];                     b0.y = q0[1];
    b1.x = q0[(size_t)16 * K];        b1.y = q0[(size_t)16 * K + 1];
    b2.x = q0[(size_t)32 * K];        b2.y = q0[(size_t)32 * K + 1];
    b3.x = q0[(size_t)48 * K];        b3.y = q0[(size_t)48 * K + 1];
    acc0 = wmma_f32(a, b0, acc0);
    acc1 = wmma_f32(a, b1, acc1);
    acc2 = wmma_f32(a, b2, acc2);
    acc3 = wmma_f32(a, b3, acc3);
  }
  // C/D layout: VGPR r holds M = r + 8*(lane>>4), N = lane&15
  const int rb = (lane >> 4) << 3;
  for (int r = 0; r < 8; ++r) {
    float v0 = acc0[r], v1 = acc1[r], v2 = acc2[r], v3 = acc3[r];
    if (relu) {
      v0 = fmaxf(v0, 0.0f); v1 = fmaxf(v1, 0.0f);
      v2 = fmaxf(v2, 0.0f); v3 = fmaxf(v3, 0.0f);
    }
    float* c = C + (size_t)(tileM + rb + r) * (size_t)N + tileN0 + lm;
    c[0] = v0; c[16] = v1; c[32] = v2; c[48] = v3;
  }
}

// ---------------------------------------------------------------------------
// Attention: one workgroup (4 waves) per (b, h, 16-row L tile).
// Full 16 x S score strip lives in LDS (128 KB of the WGP's 320 KB).
//   phase 1: scores = (Q Kt)/sqrt(hd) * attn_factor  (WMMA; K tiles via TDM)
//   phase 2: row softmax over S in LDS (8 threads per row)
//   phase 3: O = P V (WMMA; V tiles via TDM), cross-wave ds_add_f32 reduce
// ---------------------------------------------------------------------------
__global__ void attention_kernel(const float* __restrict__ Qb,
                                 const float* __restrict__ Kbuf,
                                 const float* __restrict__ Vbuf,
                                 const float* __restrict__ AF,
                                 float* __restrict__ Ob) {
  extern __shared__ float smem[];
  float* sc  = smem;                   // [16][TL_S]   score strip
  float* qt  = sc + 16 * TL_S;         // [16][32]     Q tile
  float* oa  = qt + 512;               // [16][32]     O accumulator
  float* red = oa + 512;               // [16][8]      reduction scratch
  float* st  = red + 128;              // [16]         row max / inv-sum
  float* stg = st + 16;                // [4 waves][2][16*32] TDM staging

  const int tid  = threadIdx.x;
  const int lane = tid & 31;
  const int wv   = tid >> 5;
  const int lm   = lane & 15;
  const int kk   = (lane >> 4) << 1;

  const int ltile = blockIdx.x & 127;        // L/16 = 128
  const int h     = (blockIdx.x >> 7) & 7;
  const int b     = blockIdx.x >> 10;
  const int l0    = ltile << 4;

  const float* Qp  = Qb   + (size_t)b * TL_L * TL_D + h * TL_HD;
  const float* Kp  = Kbuf + (size_t)b * TL_S * TL_D + h * TL_HD;
  const float* Vp  = Vbuf + (size_t)b * TL_S * TL_D + h * TL_HD;
  const float* afp = AF   + ((size_t)b * TL_L + l0) * TL_S;
  float*       Op  = Ob   + (size_t)b * TL_L * TL_D + h * TL_HD;

  float* mystg = stg + wv * 1024;      // this wave's double buffer

  for (int i = tid; i < 512; i += 128) {
    int r = i >> 5, c = i & 31;
    qt[i] = Qp[(size_t)(l0 + r) * TL_D + c];
    oa[i] = 0.0f;
  }
  __syncthreads();

  const float scale = 0.17677669529663687f;  // 1/sqrt(32)
  const int nst = TL_S / 16;                 // 128 S-tiles

  // ---- phase 1: score tiles (each wave strides over S tiles) ----
#ifdef TL_HAS_TDM
  {
    int buf = 0;
    tdm_load_tile_16x32(mystg, Kp + (size_t)(wv << 4) * TL_D);
    for (int stile = wv; stile < nst; stile += 4) {
      const int s0 = stile << 4;
      const int nxt = stile + 4;
      if (nxt < nst) {
        tdm_load_tile_16x32(mystg + ((buf ^ 1) << 9),
                            Kp + (size_t)(nxt << 4) * TL_D);
        tl_wait_tensorcnt1();            // current buffer is ready
      } else {
        tl_wait_tensorcnt0();
      }
      const float* kt = mystg + (buf << 9);
      v8f acc = {};
      for (int k = 0; k < TL_HD; k += 4) {
        const int ka = k + kk;
        v2f a, bb;
        a.x  = qt[lm * 32 + ka];  a.y  = qt[lm * 32 + ka + 1];
        bb.x = kt[lm * 32 + ka];  bb.y = kt[lm * 32 + ka + 1];
        acc = wmma_f32(a, bb, acc);
      }
      const int rb = (lane >> 4) << 3;
      for (int r = 0; r < 8; ++r) {
        const int mm = rb + r;
        sc[mm * TL_S + s0 + lm] =
            acc[r] * scale * afp[(size_t)mm * TL_S + s0 + lm];
      }
      buf ^= 1;
    }
  }
#else
  for (int stile = wv; stile < nst; stile += 4) {
    const int s0 = stile << 4;
    v8f acc = {};
    for (int k = 0; k < TL_HD; k += 4) {
      const int ka = k + kk;
      v2f a, bb;
      a.x = qt[lm * 32 + ka];
      a.y = qt[lm * 32 + ka + 1];
      const float* kp = Kp + (size_t)(s0 + lm) * TL_D + ka;
      bb.x = kp[0]; bb.y = kp[1];
      acc = wmma_f32(a, bb, acc);
    }
    const int rb = (lane >> 4) << 3;
    for (int r = 0; r < 8; ++r) {
      const int mm = rb + r;
      sc[mm * TL_S + s0 + lm] =
          acc[r] * scale * afp[(size_t)mm * TL_S + s0 + lm];
    }
  }
#endif
  __syncthreads();

  // ---- phase 2: softmax over S, 8 threads per row ----
  {
    const int row = tid >> 3;
    const int sub = tid & 7;
    float* rp = sc + row * TL_S;
    const int c0 = sub * (TL_S / 8);
    float mx = -3.4e38f;
    for (int c = 0; c < TL_S / 8; ++c) mx = fmaxf(mx, rp[c0 + c]);
    red[row * 8 + sub] = mx;
    __syncthreads();
    if (sub == 0) {
      float m2 = red[row * 8];
      for (int j = 1; j < 8; ++j) m2 = fmaxf(m2, red[row * 8 + j]);
      st[row] = m2;
    }
    __syncthreads();
    const float rm = st[row];
    float sum = 0.0f;
    for (int c = 0; c < TL_S / 8; ++c) {
      float e = __expf(rp[c0 + c] - rm);
      rp[c0 + c] = e;
      sum += e;
    }
    red[row * 8 + sub] = sum;
    __syncthreads();
    if (sub == 0) {
      float t = 0.0f;
      for (int j = 0; j < 8; ++j) t += red[row * 8 + j];
      st[row] = 1.0f / t;
    }
    __syncthreads();
    const float inv = st[row];
    for (int c = 0; c < TL_S / 8; ++c) rp[c0 + c] *= inv;
  }
  __syncthreads();

  // ---- phase 3: O = P @ V, per-wave partials, ds_add_f32 reduce ----
  v8f o0 = {}, o1 = {};
#ifdef TL_HAS_TDM
  {
    int buf = 0;
    tdm_load_tile_16x32(mystg, Vp + (size_t)(wv << 4) * TL_D);
    for (int stile = wv; stile < nst; stile += 4) {
      const int s0 = stile << 4;
      const int nxt = stile + 4;
      if (nxt < nst) {
        tdm_load_tile_16x32(mystg + ((buf ^ 1) << 9),
                            Vp + (size_t)(nxt << 4) * TL_D);
        tl_wait_tensorcnt1();
      } else {
        tl_wait_tensorcnt0();
      }
      const float* vt = mystg + (buf << 9);
      for (int ks = 0; ks < 4; ++ks) {
        const int kl = (ks << 2) + kk;   // local S index within tile
        v2f a, b0, b1;
        a.x = sc[lm * TL_S + s0 + kl];
        a.y = sc[lm * TL_S + s0 + kl + 1];
        b0.x = vt[kl * 32 + lm];        b0.y = vt[(kl + 1) * 32 + lm];
        b1.x = vt[kl * 32 + 16 + lm];   b1.y = vt[(kl + 1) * 32 + 16 + lm];
        o0 = wmma_f32(a, b0, o0);
        o1 = wmma_f32(a, b1, o1);
      }
      buf ^= 1;
    }
  }
#else
  for (int stile = wv; stile < nst; stile += 4) {
    const int s0 = stile << 4;
    for (int ks = 0; ks < 4; ++ks) {
      const int kb2 = s0 + (ks << 2) + kk;
      v2f a, b0, b1;
      a.x = sc[lm * TL_S + kb2];
      a.y = sc[lm * TL_S + kb2 + 1];
      const float* v0 = Vp + (size_t)kb2 * TL_D;
      const float* v1 = Vp + (size_t)(kb2 + 1) * TL_D;
      b0.x = v0[lm];       b0.y = v1[lm];
      b1.x = v0[16 + lm];  b1.y = v1[16 + lm];
      o0 = wmma_f32(a, b0, o0);
      o1 = wmma_f32(a, b1, o1);
    }
  }
#endif
  {
    const int rb = (lane >> 4) << 3;
    for (int r = 0; r < 8; ++r) {
      atomicAdd(&oa[(rb + r) * 32 + lm],      o0[r]);   // ds_add_f32
      atomicAdd(&oa[(rb + r) * 32 + 16 + lm], o1[r]);
    }
  }
  __syncthreads();
  for (int i = tid; i < 512; i += 128) {
    int r = i >> 5, c = i & 31;
    Op[(size_t)(l0 + r) * TL_D + c] = oa[i];
  }
}

// ---------------------------------------------------------------------------
// LayerNorm over D=256, one row per 256-thread block, optional residual add.
// ---------------------------------------------------------------------------
__global__ void layernorm_kernel(const float* __restrict__ in,
                                 const float* __restrict__ g,
                                 const float* __restrict__ beta,
                                 const float* __restrict__ resid,
                                 float* __restrict__ out) {
  __shared__ float red[TL_D];
  const int tid = threadIdx.x;
  const size_t row = blockIdx.x;
  const float v = in[row * TL_D + tid];
  red[tid] = v;
  __syncthreads();
  for (int s2 = TL_D / 2; s2 > 0; s2 >>= 1) {
    if (tid < s2) red[tid] += red[tid + s2];
    __syncthreads();
  }
  const float mean = red[0] * (1.0f / TL_D);
  __syncthreads();
  const float diff = v - mean;
  red[tid] = diff * diff;
  __syncthreads();
  for (int s2 = TL_D / 2; s2 > 0; s2 >>= 1) {
    if (tid < s2) red[tid] += red[tid + s2];
    __syncthreads();
  }
  const float var = red[0] * (1.0f / TL_D);
  float y = diff * rsqrtf(var + TL_EPS) * g[tid] + beta[tid];
  if (resid) y += resid[row * TL_D + tid];
  out[row * TL_D + tid] = y;
}

// ---------------------------------------------------------------------------
extern "C" void kernel_launch(void* const* d_in, const int* in_sizes, int n_in,
                              void* d_out, int out_size, void* d_ws,
                              size_t ws_size, hipStream_t stream) {
  (void)in_sizes; (void)n_in; (void)out_size; (void)ws_size;
  const float* x   = (const float*)d_in[0];   // [B,L,D]
  const float* src = (const float*)d_in[1];   // [B,S,D]
  const float* af  = (const float*)d_in[2];   // [B,L,S]
  const float* Wq  = (const float*)d_in[3];
  const float* Wk  = (const float*)d_in[4];
  const float* Wv  = (const float*)d_in[5];
  const float* Wm  = (const float*)d_in[6];
  const float* W1  = (const float*)d_in[7];   // [2D,2D]
  const float* W2  = (const float*)d_in[8];   // [D,2D]
  const float* g1  = (const float*)d_in[9];
  const float* b1  = (const float*)d_in[10];
  const float* g2  = (const float*)d_in[11];
  const float* b2  = (const float*)d_in[12];
  float* out = (float*)d_out;

  const int M = TL_B * TL_L;  // 4096 rows
  float* ws   = (float*)d_ws;
  float* Q    = ws;
  float* Kb   = Q    + (size_t)M * TL_D;
  float* Vb   = Kb   + (size_t)M * TL_D;
  float* O    = Vb   + (size_t)M * TL_D;
  float* M1   = O    + (size_t)M * TL_D;
  float* M1LN = M1   + (size_t)M * TL_D;
  float* Hb   = M1LN + (size_t)M * TL_D;       // [M, 2D]
  float* M2   = Hb   + (size_t)M * 2 * TL_D;

  const dim3 blk(128);                          // 4 waves / block
  const dim3 g256(TL_D / 64, M / 64);           // N=256 GEMMs (16x64 per wave)
  const dim3 g512(2 * TL_D / 64, M / 64);       // N=512 GEMM

  // QKV projections: [4096,256] = A[4096,256] @ W[256,256]^T
  gemm_nt_kernel<<<g256, blk, 0, stream>>>(x,   x,   Wq, Q,  TL_D, TL_D, TL_D, 0);
  gemm_nt_kernel<<<g256, blk, 0, stream>>>(src, src, Wk, Kb, TL_D, TL_D, TL_D, 0);
  gemm_nt_kernel<<<g256, blk, 0, stream>>>(src, src, Wv, Vb, TL_D, TL_D, TL_D, 0);

  // Attention: 2048 workgroups, 128 threads, ~149 KB dynamic LDS each
  const size_t smem = (size_t)TL_SMEM_FLOATS * sizeof(float);
  attention_kernel<<<dim3(TL_B * TL_H * (TL_L / 16)), blk, smem, stream>>>(
      Q, Kb, Vb, af, O);

  // message = O @ Wm^T ; LayerNorm1
  gemm_nt_kernel<<<g256, blk, 0, stream>>>(O, O, Wm, M1, TL_D, TL_D, TL_D, 0);
  layernorm_kernel<<<dim3(M), dim3(TL_D), 0, stream>>>(M1, g1, b1, nullptr, M1LN);

  // h = relu(concat(x, message) @ W1^T)  -- concat via K-split, no copy
  gemm_nt_kernel<<<g512, blk, 0, stream>>>(x, M1LN, W1, Hb, 2 * TL_D, 2 * TL_D,
                                           TL_D, 1);
  // message = h @ W2^T ; out = x + LayerNorm2(message)
  gemm_nt_kernel<<<g256, blk, 0, stream>>>(Hb, Hb, W2, M2, TL_D, 2 * TL_D,
                                           2 * TL_D, 0);
  layernorm_kernel<<<dim3(M), dim3(TL_D), 0, stream>>>(M2, g2, b2, x, out);
}